// Classical_autoencoder_11622181503576
// MI455X (gfx1250) — compile-verified
//
#include <hip/hip_runtime.h>

#define IMGSZ 512
#define BS 32
#define OH 255
#define OW 255
#define NP (OH * OW)          // 65025 patches per image
#define CEPS 1e-8f
#define TPW 8                 // 16-patch tiles per wave (amortizes weight setup)

typedef float v2f __attribute__((ext_vector_type(2)));
typedef float v8f __attribute__((ext_vector_type(8)));

// ---------------------------------------------------------------------------
// Kernel 1: per-patch MLP + cosine similarity, one 16-patch tile per WMMA.
//
// Trick: the f32 A(16x4) layout (lane=M, reg r/half h -> K=2h+r) and B(4x16)
// layout (lane=N, reg/half -> K) carry identical per-lane registers for a
// matrix and its transpose. So the gathered X fragments (lane n = patch n,
// float2 of features 4c+2h..+1) serve as B = X^T, and A = W1^T duplicated
// into rows 0-3 AND 8-11. Then D regs 0..3 hold h1_pre[patch n, q] in EVERY
// lane of both halves -> layers 2/3 + cosine are fully lane-local; only 3
// shfl_xor(16) combine the half-wave feature splits. No LDS, no transposes.
// ---------------------------------------------------------------------------
__global__ __launch_bounds__(256) void sim_kernel(
    const float* __restrict__ img,
    const float* __restrict__ W1, const float* __restrict__ b1,
    const float* __restrict__ W2, const float* __restrict__ b2,
    const float* __restrict__ W3, const float* __restrict__ b3,
    float* __restrict__ simbuf)
{
    const int lane = threadIdx.x & 31;
    const int h    = lane >> 4;          // half: feature columns kc = 2h, 2h+1
    const int n    = lane & 15;          // patch index within tile
    const int wave = threadIdx.x >> 5;
    const int b    = blockIdx.y;
    const int pw   = blockIdx.x * (128 * TPW / 8) * 8 + wave * (16 * TPW);
    if (pw >= NP) return;                // wave-uniform

    const float* imgb = img + (size_t)b * IMGSZ * IMGSZ;

    // A operand: W1^T duplicated into rows {0-3, 8-11}; rows {4-7,12-15}=0.
    // A[m, f] at (reg r, half h, lane m) with f = 4c + 2h + r.
    v2f aw[4];
    {
        const bool valid = (n < 4) | (n >= 8 && n < 12);
        const int  q = n & 3;
        #pragma unroll
        for (int c = 0; c < 4; ++c)
            #pragma unroll
            for (int r = 0; r < 2; ++r)
                aw[c][r] = valid ? W1[(4 * c + 2 * h + r) * 4 + q] : 0.0f;
    }
    // C: bias b1 on rows 0-3 (and 8-11 via reg replication across halves)
    const v8f cbias = { b1[0], b1[1], b1[2], b1[3], 0.f, 0.f, 0.f, 0.f };

    // Wave-uniform W2/b2 (SGPRs) + per-lane slices of W3/b3 for the lane's
    // 8 local features f = 4c + 2h + r.
    float w2[16], b2v[4], w3l[4][2][4], b3l[4][2];
    #pragma unroll
    for (int t = 0; t < 16; ++t) w2[t] = W2[t];
    #pragma unroll
    for (int k = 0; k < 4; ++k)  b2v[k] = b2[k];
    #pragma unroll
    for (int c = 0; c < 4; ++c)
        #pragma unroll
        for (int r = 0; r < 2; ++r) {
            const int f = 4 * c + 2 * h + r;
            b3l[c][r] = b3[f];
            #pragma unroll
            for (int k = 0; k < 4; ++k) w3l[c][r][k] = W3[k * 16 + f];
        }

    for (int t = 0; t < TPW; ++t) {
        const int p0 = pw + t * 16;
        if (p0 >= NP) break;             // uniform: EXEC stays all-1s for WMMA

        // B operand = X^T: lane n holds patch p0+n, features (kr=c, kc=2h+r)
        // -> one contiguous float2 per kernel row.
        v2f x[4];
        {
            const int p = min(p0 + n, NP - 1);
            const int i = p / OW, j = p - i * OW;
            const float* rb = imgb + (size_t)(2 * i) * IMGSZ + 2 * j + 2 * h;
            #pragma unroll
            for (int c = 0; c < 4; ++c)
                x[c] = *(const v2f*)(rb + c * IMGSZ);
        }

        v8f acc = cbias;
        #pragma unroll
        for (int c = 0; c < 4; ++c)
            acc = __builtin_amdgcn_wmma_f32_16x16x4_f32(false, aw[c], false, x[c],
                                                        (short)0, acc, false, false);

        // acc[q], q=0..3: h1_pre[patch n, q] replicated in all 32 lanes.
        float h1q[4];
        #pragma unroll
        for (int q = 0; q < 4; ++q) h1q[q] = fmaxf(acc[q], 0.0f);

        float h2[4];
        #pragma unroll
        for (int k = 0; k < 4; ++k) {
            float s = b2v[k];
            #pragma unroll
            for (int q = 0; q < 4; ++q) s = fmaf(h1q[q], w2[q * 4 + k], s);
            h2[k] = fmaxf(s, 0.0f);
        }

        // Layer 3 + cosine terms, only for this lane's 8 features.
        float dot = 0.0f, nx2 = 0.0f, ny2 = 0.0f;
        #pragma unroll
        for (int c = 0; c < 4; ++c)
            #pragma unroll
            for (int r = 0; r < 2; ++r) {
                float yv = b3l[c][r];
                #pragma unroll
                for (int k = 0; k < 4; ++k) yv = fmaf(h2[k], w3l[c][r][k], yv);
                yv = fmaxf(yv, 0.0f);
                const float xv = x[c][r];
                dot = fmaf(xv, yv, dot);
                nx2 = fmaf(xv, xv, nx2);
                ny2 = fmaf(yv, yv, ny2);
            }

        // Combine the two half-wave feature sets (kc 0,1 vs kc 2,3).
        dot += __shfl_xor(dot, 16, 32);
        nx2 += __shfl_xor(nx2, 16, 32);
        ny2 += __shfl_xor(ny2, 16, 32);

        const float sim = dot / (fmaxf(sqrtf(nx2), CEPS) * fmaxf(sqrtf(ny2), CEPS));

        if (h == 0 && p0 + n < NP)       // 16-lane coalesced store
            simbuf[(size_t)b * NP + p0 + n] = sim;
    }
}

// ---------------------------------------------------------------------------
// Kernel 2: overlap-average fold. For y in {2Y, 2Y+1} the covering patch-row
// range is identically [max(0,Y-1), min(254,Y)] (same for x), so the output
// is constant on 2x2 blocks: compute once, write 4 pixels as two float2.
// ---------------------------------------------------------------------------
__global__ __launch_bounds__(256) void fold_kernel(const float* __restrict__ sim,
                                                   float* __restrict__ out)
{
    const int idx = blockIdx.x * blockDim.x + threadIdx.x;
    if (idx >= BS * 256 * 256) return;
    const int X = idx & 255;
    const int Y = (idx >> 8) & 255;
    const int b = idx >> 16;

    const int i0 = max(0, Y - 1), i1 = min(OH - 1, Y);
    const int j0 = max(0, X - 1), j1 = min(OW - 1, X);

    const float* s = sim + (size_t)b * NP;
    float acc = 0.0f;
    for (int i = i0; i <= i1; ++i)
        for (int j = j0; j <= j1; ++j)
            acc += s[i * OW + j];
    const float v = acc / (float)((i1 - i0 + 1) * (j1 - j0 + 1));

    v2f vv = { v, v };
    float* o = out + (size_t)b * IMGSZ * IMGSZ + (size_t)(2 * Y) * IMGSZ + 2 * X;
    *(v2f*)o = vv;
    *(v2f*)(o + IMGSZ) = vv;
}

extern "C" void kernel_launch(void* const* d_in, const int* in_sizes, int n_in,
                              void* d_out, int out_size, void* d_ws, size_t ws_size,
                              hipStream_t stream)
{
    const float* img = (const float*)d_in[0];
    const float* W1  = (const float*)d_in[1];
    const float* b1  = (const float*)d_in[2];
    const float* W2  = (const float*)d_in[3];
    const float* b2  = (const float*)d_in[4];
    const float* W3  = (const float*)d_in[5];
    const float* b3  = (const float*)d_in[6];
    float* simbuf = (float*)d_ws;                    // BS*NP floats = 8.3 MB (L2-resident)
    float* out    = (float*)d_out;

    // 8 waves/block, each wave runs TPW=8 tiles (128 patches) -> 1024/block
    dim3 g1((NP + 128 * TPW - 1) / (128 * TPW), BS);
    sim_kernel<<<g1, 256, 0, stream>>>(img, W1, b1, W2, b2, W3, b3, simbuf);

    const int nfold = BS * 256 * 256;                // one thread per 2x2 output block
    fold_kernel<<<(nfold + 255) / 256, 256, 0, stream>>>(simbuf, out);
}